// CocoaLoss_30030411333999
// MI455X (gfx1250) — compile-verified
//
#include <hip/hip_runtime.h>
#include <math.h>
#include <stdint.h>

typedef __attribute__((ext_vector_type(2))) float v2f;
typedef __attribute__((ext_vector_type(8))) float v8f;

#define B_N 4096
#define D_N 4096
#define L_N 64
#define INV_TAU 10.0f   // 1/0.1
#define THRESH 32       // L/2
#define LDS_STRIDE 36   // 16B-aligned rows, bank-conflict-free operand reads

// ---------------- workspace layout (floats) ----------------
// wsf[0] : pos_sum   (float, atomic)
// wsf[1] : neg_sum   (float, atomic; x-gram + y-gram combined, LAM=1)
// wsf[2] : n_neg     (int,   atomic)
// wsf[3] : pad
// wsf[4            .. 4+B)   : invnorm_x
// wsf[4+B          .. 4+2B)  : invnorm_y
// wsf[4+2B         .. 4+3B)  : neg mask (int)

__global__ void zero_acc_kernel(float* wsf) {
    if (threadIdx.x < 4) wsf[threadIdx.x] = 0.0f;
}

// One 256-thread block per row: ||x||, ||y||, <x,y>  ->  invnorms + pos term.
__global__ __launch_bounds__(256)
void rowstats_kernel(const float* __restrict__ x, const float* __restrict__ y,
                     float* __restrict__ invnx, float* __restrict__ invny,
                     float* __restrict__ pos_acc) {
    __shared__ float sxx[256], syy[256], sxy[256];
    const int row = blockIdx.x;
    const float4* xr = (const float4*)(x + (size_t)row * D_N);
    const float4* yr = (const float4*)(y + (size_t)row * D_N);
    float axx = 0.f, ayy = 0.f, axy = 0.f;
    for (int k = threadIdx.x; k < D_N / 4; k += 256) {
        float4 xv = xr[k], yv = yr[k];
        axx += xv.x*xv.x + xv.y*xv.y + xv.z*xv.z + xv.w*xv.w;
        ayy += yv.x*yv.x + yv.y*yv.y + yv.z*yv.z + yv.w*yv.w;
        axy += xv.x*yv.x + xv.y*yv.y + xv.z*yv.z + xv.w*yv.w;
    }
    sxx[threadIdx.x] = axx; syy[threadIdx.x] = ayy; sxy[threadIdx.x] = axy;
    __syncthreads();
    for (int off = 128; off > 0; off >>= 1) {
        if (threadIdx.x < off) {
            sxx[threadIdx.x] += sxx[threadIdx.x + off];
            syy[threadIdx.x] += syy[threadIdx.x + off];
            sxy[threadIdx.x] += sxy[threadIdx.x + off];
        }
        __syncthreads();
    }
    if (threadIdx.x == 0) {
        float inx = rsqrtf(sxx[0]);
        float iny = rsqrtf(syy[0]);
        invnx[row] = inx;
        invny[row] = iny;
        float c = sxy[0] * inx * iny;
        atomicAdd(pos_acc, __expf((1.0f - c) * INV_TAU));
    }
}

// One wave (32 lanes) per row of labels: count zeros, mask = zeros > THRESH.
__global__ void mask_kernel(const int* __restrict__ labels,
                            int* __restrict__ mask, int* __restrict__ nneg) {
    const int row = blockIdx.x;
    const int t = threadIdx.x;
    const int* lr = labels + (size_t)row * L_N;
    int z = (lr[t] == 0 ? 1 : 0) + (lr[t + 32] == 0 ? 1 : 0);
    for (int off = 16; off > 0; off >>= 1) z += __shfl_down(z, off);
    if (t == 0) {
        int m = (z > THRESH) ? 1 : 0;
        mask[row] = m;
        if (m) atomicAdd(nneg, 1);
    }
}

// Async DMA of one 64x32 f32 block straight into LDS (no VGPR round trip).
// Tracked by ASYNCcnt; caller does s_wait_asynccnt + barrier.
__device__ __forceinline__
void async_stage_b128(const float* __restrict__ gsrc, const float* lbase,
                      int tid, int k0) {
#pragma unroll
    for (int q = 0; q < 4; ++q) {
        int idx = tid + q * 128;                 // 512 float4 per 64x32 block
        int r = idx >> 3, c = (idx & 7) * 4;
        // low 32 bits of a flat LDS address are the DS byte offset
        unsigned la = (unsigned)(uintptr_t)(lbase + r * LDS_STRIDE + c);
        const float* ga = gsrc + (size_t)r * D_N + k0 + c;
        asm volatile("global_load_async_to_lds_b128 %0, %1, off"
                     :: "v"(la), "v"(ga) : "memory");
    }
}

__device__ __forceinline__ void wait_async0() {
    asm volatile("s_wait_asynccnt 0x0" ::: "memory");
}

// Triangular tiled Gram + masked-exp reduction.
// grid = (64, 64, 2): blockIdx.x = tile col tj, blockIdx.y = tile row ti (tj>=ti),
// blockIdx.z selects x (0) or y (1). 128 threads = 4 waves, each wave owns a
// 32x32 patch = 2x2 WMMA 16x16 tiles; macro tile 64x64, K blocked by 32.
// Double-buffered LDS stages filled by GLOBAL_LOAD_ASYNC_TO_LDS_B128 so the
// stage-(k+1) DMA overlaps the stage-k WMMA work.
__global__ __launch_bounds__(128)
void gram_reduce_kernel(const float* __restrict__ x, const float* __restrict__ y,
                        const float* __restrict__ invnx, const float* __restrict__ invny,
                        const int* __restrict__ mask, float* __restrict__ neg_acc) {
    const int tj = blockIdx.x;
    const int ti = blockIdx.y;
    if (tj < ti) return;  // symmetry: upper triangle of tile-blocks only

    const float* __restrict__ Z    = blockIdx.z ? y : x;
    const float* __restrict__ invn = blockIdx.z ? invny : invnx;
    const int iBase = ti * 64;
    const int jBase = tj * 64;

    __shared__ float As[2][64 * LDS_STRIDE];
    __shared__ float Bs[2][64 * LDS_STRIDE];
    __shared__ float invI[64], invJ[64];
    __shared__ int   mI[64],  mJ[64];

    const int tid = threadIdx.x;
    if (tid < 64) { invI[tid] = invn[iBase + tid]; mI[tid] = mask[iBase + tid]; }
    else { int t = tid - 64; invJ[t] = invn[jBase + t]; mJ[t] = mask[jBase + t]; }

    const bool diag = (ti == tj);
    const float* Asrc = Z + (size_t)iBase * D_N;
    const float* Bsrc = Z + (size_t)jBase * D_N;

    const int w    = tid >> 5;
    const int lane = tid & 31;
    const int wRow = (w >> 1) * 32;   // wave's 32-row patch inside 64
    const int wCol = (w & 1) * 32;    // wave's 32-col patch inside 64
    const int mA   = lane & 15;       // A/B fragment: matrix index within tile
    const int koff = (lane >> 4) * 2; // half-wave K split {0,1} vs {2,3}

    v8f acc[2][2] = {};

    // Prologue: fill stage 0.
    async_stage_b128(Asrc, As[0], tid, 0);
    if (!diag) async_stage_b128(Bsrc, Bs[0], tid, 0);
    wait_async0();
    __syncthreads();

    for (int k0 = 0; k0 < D_N; k0 += 32) {
        const int cur = (k0 >> 5) & 1;

        // Kick off the next stage's DMA before computing this one.
        if (k0 + 32 < D_N) {
            async_stage_b128(Asrc, As[cur ^ 1], tid, k0 + 32);
            if (!diag) async_stage_b128(Bsrc, Bs[cur ^ 1], tid, k0 + 32);
        }

        const float* Ac = As[cur];
        const float* Bc = diag ? As[cur] : Bs[cur];

        #pragma unroll
        for (int kk = 0; kk < 32; kk += 4) {
            const float* a0p = &Ac[(wRow + mA) * LDS_STRIDE + kk + koff];
            const float* a1p = a0p + 16 * LDS_STRIDE;
            const float* b0p = &Bc[(wCol + mA) * LDS_STRIDE + kk + koff];
            const float* b1p = b0p + 16 * LDS_STRIDE;
            v2f a0 = {a0p[0], a0p[1]};
            v2f a1 = {a1p[0], a1p[1]};
            v2f b0 = {b0p[0], b0p[1]};
            v2f b1 = {b1p[0], b1p[1]};
            acc[0][0] = __builtin_amdgcn_wmma_f32_16x16x4_f32(
                false, a0, false, b0, (short)0, acc[0][0], false, false);
            acc[0][1] = __builtin_amdgcn_wmma_f32_16x16x4_f32(
                false, a0, false, b1, (short)0, acc[0][1], false, false);
            acc[1][0] = __builtin_amdgcn_wmma_f32_16x16x4_f32(
                false, a1, false, b0, (short)0, acc[1][0], false, false);
            acc[1][1] = __builtin_amdgcn_wmma_f32_16x16x4_f32(
                false, a1, false, b1, (short)0, acc[1][1], false, false);
        }

        // Retire this wave's async DMAs into the next stage, then make all
        // waves' LDS writes visible. (ds_loads of `cur` were already forced
        // complete by the WMMA issue waits, so reusing `cur` next iter is safe.)
        wait_async0();
        __syncthreads();
    }

    // Epilogue: C/D layout — VGPR r holds (m = r + 8*(lane>=16), n = lane&15).
    // Unordered-pair weight: (mask_i XOR mask_j) counted once for j > i.
    float local = 0.0f;
    const int nSub = lane & 15;
    const int rAdd = (lane >> 4) << 3;
    #pragma unroll
    for (int tm = 0; tm < 2; ++tm)
        #pragma unroll
        for (int tn = 0; tn < 2; ++tn)
            #pragma unroll
            for (int r = 0; r < 8; ++r) {
                int m = wRow + tm * 16 + r + rAdd;
                int n = wCol + tn * 16 + nSub;
                if ((jBase + n) > (iBase + m) && (mI[m] ^ mJ[n])) {
                    float sim = acc[tm][tn][r] * invI[m] * invJ[n];
                    local += __expf(sim * INV_TAU);
                }
            }
    for (int off = 16; off > 0; off >>= 1) local += __shfl_down(local, off);
    if (lane == 0) atomicAdd(neg_acc, local);
}

__global__ void finalize_kernel(const float* __restrict__ wsf, float* __restrict__ out) {
    float pos = wsf[0];
    float neg = wsf[1];
    int nneg = ((const int*)wsf)[2];
    long long cnt = (long long)nneg * (long long)(B_N - nneg);
    float r = pos * (1.0f / (float)B_N);
    if (cnt > 0) r += neg / (float)cnt;   // LAM = 1, same cnt for x and y grams
    out[0] = r;
}

extern "C" void kernel_launch(void* const* d_in, const int* in_sizes, int n_in,
                              void* d_out, int out_size, void* d_ws, size_t ws_size,
                              hipStream_t stream) {
    const float* x      = (const float*)d_in[0];
    const float* y      = (const float*)d_in[1];
    const int*   labels = (const int*)d_in[2];

    float* wsf   = (float*)d_ws;
    float* invnx = wsf + 4;
    float* invny = wsf + 4 + B_N;
    int*   mask  = (int*)(wsf + 4 + 2 * B_N);

    zero_acc_kernel<<<1, 32, 0, stream>>>(wsf);
    rowstats_kernel<<<B_N, 256, 0, stream>>>(x, y, invnx, invny, &wsf[0]);
    mask_kernel<<<B_N, 32, 0, stream>>>(labels, mask, (int*)&wsf[2]);
    dim3 grid(64, 64, 2);
    gram_reduce_kernel<<<grid, 128, 0, stream>>>(x, y, invnx, invny, mask, &wsf[1]);
    finalize_kernel<<<1, 1, 0, stream>>>(wsf, (float*)d_out);
}